// GCN_80616536146466
// MI455X (gfx1250) — compile-verified
//
#include <hip/hip_runtime.h>
#include <math.h>

typedef __attribute__((ext_vector_type(2))) float v2f;
typedef __attribute__((ext_vector_type(8))) float v8f;

// ---------------------------------------------------------------------------
// Normalization: deg = segment_sum(ew over col); dinv = rsqrt(deg) (0 if <=0);
// norm[e] = dinv[row]*ew*dinv[col]
// ---------------------------------------------------------------------------
__global__ void deg_kernel(const int* __restrict__ col, const float* __restrict__ ew,
                           float* __restrict__ deg, int E) {
    int e = blockIdx.x * blockDim.x + threadIdx.x;
    if (e < E) atomicAdd(&deg[col[e]], ew[e]);
}

__global__ void dinv_kernel(float* __restrict__ deg, int N) {
    int n = blockIdx.x * blockDim.x + threadIdx.x;
    if (n < N) {
        float d = deg[n];
        deg[n] = (d > 0.0f) ? rsqrtf(d) : 0.0f;
    }
}

__global__ void norm_kernel(const int* __restrict__ row, const int* __restrict__ col,
                            const float* __restrict__ ew, const float* __restrict__ dinv,
                            float* __restrict__ norm, int E) {
    int e = blockIdx.x * blockDim.x + threadIdx.x;
    if (e < E) norm[e] = dinv[row[e]] * ew[e] * dinv[col[e]];
}

// ---------------------------------------------------------------------------
// WMMA GEMM:  C[n][m] = sum_k (A[n][k] (+ bin[k])) * W[m][k]
//   A: nrows x K row-major, W: NOUT x K row-major, C: nrows x NOUT
// One wave computes a 16-row M-tile across all NT = ceil(NOUT/16) N-tiles,
// reusing the A fragment. f32 WMMA 16x16x4 -> exact fp32 semantics.
// ---------------------------------------------------------------------------
template <int K, int NOUT, bool HASB>
__global__ __launch_bounds__(128)
void gemm_wmma_f32(const float* __restrict__ A, const float* __restrict__ W,
                   const float* __restrict__ bin, float* __restrict__ C, int nrows) {
    constexpr int NT = (NOUT + 15) / 16;

    const int wave  = threadIdx.x >> 5;          // 0..3
    const int lane  = threadIdx.x & 31;
    const int mtile = blockIdx.x * 4 + wave;
    const int row0  = mtile * 16;
    if (row0 >= nrows) return;                   // wave-uniform exit (no WMMA on this wave)

    const int lmod  = lane & 15;                 // M (for A) / N (for B,C,D) within tile
    const int khalf = (lane >> 4) * 2;           // lanes 16-31 hold K = k+2, k+3

    int arow = row0 + lmod;
    if (arow >= nrows) arow = nrows - 1;         // clamp (loads only; stores are guarded)
    const float* __restrict__ Arow = A + (size_t)arow * K;

    v8f acc[NT] = {};

    for (int k = 0; k < K; k += 4) {
        v2f a;
        if (HASB) {
            a.x = Arow[k + khalf + 0] + bin[k + khalf + 0];
            a.y = Arow[k + khalf + 1] + bin[k + khalf + 1];
        } else {
            a.x = Arow[k + khalf + 0];
            a.y = Arow[k + khalf + 1];
        }
#pragma unroll
        for (int t = 0; t < NT; ++t) {
            const int ncol = t * 16 + lmod;
            v2f b;
            if (ncol < NOUT) {                   // reconverges before WMMA (EXEC all-ones)
                const float* __restrict__ Wrow = W + (size_t)ncol * K;
                b.x = Wrow[k + khalf + 0];
                b.y = Wrow[k + khalf + 1];
            } else {
                b.x = 0.0f;
                b.y = 0.0f;
            }
            acc[t] = __builtin_amdgcn_wmma_f32_16x16x4_f32(
                false, a, false, b, (short)0, acc[t], false, false);
        }
    }

    // D layout: VGPR r = row M=r (lanes 0-15) / M=r+8 (lanes 16-31), N = lmod
    const int mofs = (lane >> 4) * 8;
#pragma unroll
    for (int t = 0; t < NT; ++t) {
        const int ncol = t * 16 + lmod;
        if (ncol >= NOUT) continue;
#pragma unroll
        for (int r = 0; r < 8; ++r) {
            const int m = row0 + mofs + r;
            if (m < nrows) C[(size_t)m * NOUT + ncol] = acc[t][r];
        }
    }
}

// ---------------------------------------------------------------------------
// Edge aggregation (H=64): 16 lanes per edge, float4 gather + 4 atomic adds.
// agg[col[e]][:] += h[row[e]][:] * norm[e]
// ---------------------------------------------------------------------------
__global__ void aggregate64(const float* __restrict__ h, const int* __restrict__ row,
                            const int* __restrict__ col, const float* __restrict__ norm,
                            float* __restrict__ agg, int E) {
    long long i = (long long)blockIdx.x * blockDim.x + threadIdx.x;
    long long total = (long long)E * 16;
    if (i >= total) return;
    const int e  = (int)(i >> 4);
    const int f4 = (int)(i & 15) * 4;
    const int r = row[e], c = col[e];
    const float w = norm[e];
    const float4 v = *(const float4*)(h + (size_t)r * 64 + f4);
    float* dst = agg + (size_t)c * 64 + f4;
    atomicAdd(dst + 0, v.x * w);
    atomicAdd(dst + 1, v.y * w);
    atomicAdd(dst + 2, v.z * w);
    atomicAdd(dst + 3, v.w * w);
}

// Edge aggregation for the 10-wide output layer: 1 thread per edge.
__global__ void aggregate10(const float* __restrict__ h, const int* __restrict__ row,
                            const int* __restrict__ col, const float* __restrict__ norm,
                            float* __restrict__ agg, int E) {
    int e = blockIdx.x * blockDim.x + threadIdx.x;
    if (e >= E) return;
    const int r = row[e], c = col[e];
    const float w = norm[e];
    const float* __restrict__ src = h + (size_t)r * 10;
    float* dst = agg + (size_t)c * 10;
#pragma unroll
    for (int f = 0; f < 10; ++f) atomicAdd(dst + f, src[f] * w);
}

// ---------------------------------------------------------------------------
// Finalize: logits = agg3 + b3 ; soft = softmax(logits)
// d_out = [logits (N*10) | soft (N*10)]
// ---------------------------------------------------------------------------
__global__ void finalize_kernel(const float* __restrict__ agg3, const float* __restrict__ b3,
                                float* __restrict__ out, int N) {
    int n = blockIdx.x * blockDim.x + threadIdx.x;
    if (n >= N) return;
    float v[10];
    float mx = -INFINITY;
#pragma unroll
    for (int i = 0; i < 10; ++i) {
        v[i] = agg3[(size_t)n * 10 + i] + b3[i];
        mx = fmaxf(mx, v[i]);
    }
    float ex[10];
    float s = 0.0f;
#pragma unroll
    for (int i = 0; i < 10; ++i) {
        ex[i] = expf(v[i] - mx);
        s += ex[i];
    }
    const float inv = 1.0f / s;
#pragma unroll
    for (int i = 0; i < 10; ++i) {
        out[(size_t)n * 10 + i] = v[i];
        out[(size_t)N * 10 + (size_t)n * 10 + i] = ex[i] * inv;
    }
}

// ---------------------------------------------------------------------------
extern "C" void kernel_launch(void* const* d_in, const int* in_sizes, int n_in,
                              void* d_out, int out_size, void* d_ws, size_t ws_size,
                              hipStream_t stream) {
    const float* x  = (const float*)d_in[0];
    const int*   ei = (const int*)d_in[1];
    const float* ew = (const float*)d_in[2];
    const float* W1 = (const float*)d_in[3];
    const float* b1 = (const float*)d_in[4];
    const float* W2 = (const float*)d_in[5];
    const float* b2 = (const float*)d_in[6];
    const float* W3 = (const float*)d_in[7];
    const float* b3 = (const float*)d_in[8];

    const int N = in_sizes[0] / 128;   // 50000
    const int E = in_sizes[2];         // 800000
    const int* row = ei;
    const int* col = ei + E;

    float* out = (float*)d_out;

    // Workspace layout (floats): deg/dinv[N] | norm[E] | t64[N*64] | agg64[N*64] | t10[N*10] | agg10[N*10]
    float* deg   = (float*)d_ws;
    float* norm  = deg + N;
    float* t64   = norm + E;
    float* agg64 = t64 + (size_t)N * 64;
    float* t10   = agg64 + (size_t)N * 64;
    float* agg10 = t10 + (size_t)N * 10;

    const int TB = 256;
    const int mtiles = (N + 15) / 16;
    const int gemmBlocks = (mtiles + 3) / 4;     // 4 waves (M-tiles) per 128-thread block
    const long long agg64Threads = (long long)E * 16;
    const int agg64Blocks = (int)((agg64Threads + TB - 1) / TB);

    // --- normalization ---
    hipMemsetAsync(deg, 0, (size_t)N * sizeof(float), stream);
    deg_kernel<<<(E + TB - 1) / TB, TB, 0, stream>>>(col, ew, deg, E);
    dinv_kernel<<<(N + TB - 1) / TB, TB, 0, stream>>>(deg, N);
    norm_kernel<<<(E + TB - 1) / TB, TB, 0, stream>>>(row, col, ew, deg, norm, E);

    // --- layer 1: t1 = x @ W1^T ; agg1 = scatter(t1) ---
    gemm_wmma_f32<128, 64, false><<<gemmBlocks, 128, 0, stream>>>(x, W1, nullptr, t64, N);
    hipMemsetAsync(agg64, 0, (size_t)N * 64 * sizeof(float), stream);
    aggregate64<<<agg64Blocks, TB, 0, stream>>>(t64, row, col, norm, agg64, E);

    // --- layer 2: t2 = (agg1 + b1) @ W2^T ; agg2 = scatter(t2) ---
    gemm_wmma_f32<64, 64, true><<<gemmBlocks, 128, 0, stream>>>(agg64, W2, b1, t64, N);
    hipMemsetAsync(agg64, 0, (size_t)N * 64 * sizeof(float), stream);
    aggregate64<<<agg64Blocks, TB, 0, stream>>>(t64, row, col, norm, agg64, E);

    // --- layer 3: t3 = (agg2 + b2) @ W3^T ; agg3 = scatter(t3) ---
    gemm_wmma_f32<64, 10, true><<<gemmBlocks, 128, 0, stream>>>(agg64, W3, b2, t10, N);
    hipMemsetAsync(agg10, 0, (size_t)N * 10 * sizeof(float), stream);
    aggregate10<<<(E + TB - 1) / TB, TB, 0, stream>>>(t10, row, col, norm, agg10, E);

    // --- logits + softmax ---
    finalize_kernel<<<(N + TB - 1) / TB, TB, 0, stream>>>(agg10, b3, out, N);
}